// CannyDetector_7842610283297
// MI455X (gfx1250) — compile-verified
//
#include <hip/hip_runtime.h>

typedef __attribute__((ext_vector_type(2))) float v2f;
typedef __attribute__((ext_vector_type(8))) float v8f;

#define Himg 1024
#define Wimg 1024
#define HWimg (Himg * Wimg)
#define NBATCH 4
#define NCH 3
#define NPIX (NBATCH * HWimg)        /* 4,194,304  */
#define NTOT (NBATCH * NCH * HWimg)  /* 12,582,912 */
#define MED_RANK 6291455u            /* (NTOT-1)/2 */

/* grad-kernel tile geometry */
#define TILE 48
#define IND 80   /* input region 80x80 covers [-7..72] around tile      */
#define INS 84   /* LDS row stride for s_in (84 % 64 != 0)              */
#define HBS 68   /* LDS row stride for s_hb / s_bl                      */
#define BLD 64   /* blurred region 64x64 covers [-1..62]                */

/* ---- async global->LDS path (gfx1250 GLOBAL_LOAD_ASYNC_TO_LDS_B32) ---- */
#if __has_builtin(__builtin_amdgcn_global_load_async_to_lds_b32) && \
    __has_builtin(__builtin_amdgcn_s_wait_asynccnt)
  #define HAVE_ASYNC 1
  #define ASYNC_CP(dst, srcp)                                              \
      __builtin_amdgcn_global_load_async_to_lds_b32(                       \
          (__attribute__((address_space(1))) int*)(srcp),                  \
          (__attribute__((address_space(3))) int*)(dst), 0, 0)
  #define WAIT_ASYNC() __builtin_amdgcn_s_wait_asynccnt(0)
#elif __has_builtin(__builtin_amdgcn_load_to_lds) && \
      __has_builtin(__builtin_amdgcn_s_wait_asynccnt)
  #define HAVE_ASYNC 1
  #define ASYNC_CP(dst, srcp)                                              \
      __builtin_amdgcn_load_to_lds(                                        \
          (void*)(srcp),                                                   \
          (__attribute__((address_space(3))) void*)(dst), 4, 0, 0)
  #define WAIT_ASYNC() __builtin_amdgcn_s_wait_asynccnt(0)
#else
  #define HAVE_ASYNC 0
#endif

// ---------------------------------------------------------------------------
// Median: radix-select on float bit patterns (all values >= 0 -> monotonic)
// ---------------------------------------------------------------------------
__global__ __launch_bounds__(256)
void hist1_kernel(const float4* __restrict__ img4, unsigned* __restrict__ hist, int n4) {
    int i = blockIdx.x * blockDim.x + threadIdx.x;
    int stride = gridDim.x * blockDim.x;
    for (; i < n4; i += stride) {
        float4 v = img4[i];
        atomicAdd(&hist[__float_as_uint(v.x) >> 16], 1u);
        atomicAdd(&hist[__float_as_uint(v.y) >> 16], 1u);
        atomicAdd(&hist[__float_as_uint(v.z) >> 16], 1u);
        atomicAdd(&hist[__float_as_uint(v.w) >> 16], 1u);
    }
}

__global__ __launch_bounds__(256)
void hist2_kernel(const float4* __restrict__ img4, const unsigned* __restrict__ info,
                  unsigned* __restrict__ hist, int n4) {
    const unsigned hi = info[0];
    int i = blockIdx.x * blockDim.x + threadIdx.x;
    int stride = gridDim.x * blockDim.x;
    for (; i < n4; i += stride) {
        float4 v = img4[i];
        unsigned b;
        b = __float_as_uint(v.x); if ((b >> 16) == hi) atomicAdd(&hist[b & 0xffffu], 1u);
        b = __float_as_uint(v.y); if ((b >> 16) == hi) atomicAdd(&hist[b & 0xffffu], 1u);
        b = __float_as_uint(v.z); if ((b >> 16) == hi) atomicAdd(&hist[b & 0xffffu], 1u);
        b = __float_as_uint(v.w); if ((b >> 16) == hi) atomicAdd(&hist[b & 0xffffu], 1u);
    }
}

__global__ __launch_bounds__(1024)
void select_kernel(const unsigned* __restrict__ hist, unsigned* __restrict__ info, int phase) {
    __shared__ unsigned part[1024];
    const unsigned t = threadIdx.x;
    unsigned s = 0;
    for (int j = 0; j < 64; ++j) s += hist[t * 64 + j];
    part[t] = s;
    __syncthreads();
    if (t == 0) {
        unsigned long long rank = (phase == 0) ? (unsigned long long)MED_RANK
                                               : (unsigned long long)info[1];
        unsigned long long cum = 0;
        int seg = 0;
        for (; seg < 1024; ++seg) {
            if (cum + part[seg] > rank) break;
            cum += part[seg];
        }
        int bin = seg * 64;
        for (int j = 0; j < 64; ++j) {
            unsigned h = hist[seg * 64 + j];
            if (cum + h > rank) { bin = seg * 64 + j; break; }
            cum += h;
        }
        if (phase == 0) {
            info[0] = (unsigned)bin;            /* selected high16 */
            info[1] = (unsigned)(rank - cum);   /* residual rank   */
        } else {
            unsigned bits = (info[0] << 16) | (unsigned)bin;
            float v = __uint_as_float(bits);    /* exact lower median */
            float t1 = fmaxf(0.0f, 0.7f * v) * 6.0f;
            float t2 = fminf(1.0f, 1.3f * v) * 6.0f;
            ((float*)info)[4] = fminf(t1, t2);  /* low  */
            ((float*)info)[5] = fmaxf(t1, t2);  /* high */
        }
    }
}

// ---------------------------------------------------------------------------
// Fused separable Gaussian blur (f32 WMMA 16x16x4) + Sobel + orientation
// ---------------------------------------------------------------------------
__global__ __launch_bounds__(256)
void grad_kernel(const float* __restrict__ img, const float* __restrict__ gauss,
                 float* __restrict__ gmag, unsigned char* __restrict__ dirm) {
    __shared__ float s_in[IND * INS];   /* input tile, global [-7..72]^2        */
    __shared__ float s_hb[IND * HBS];   /* h-blur: rows [-7..72], cols [-1..62] */
    __shared__ float s_bl[BLD * HBS];   /* blurred: [-1..62]^2                  */
    __shared__ float s_gp[48];          /* zero-padded gaussian: s_gp[d+15]=g[d] */

    const int tid   = threadIdx.x;
    const int lane  = tid & 31;
    const int wave  = tid >> 5;
    const int mrow  = lane & 15;     /* M (or N) index of this lane        */
    const int khalf = lane >> 4;     /* lanes 16-31 hold upper K / M half  */

    const int tx = blockIdx.x, ty = blockIdx.y, b = blockIdx.z;
    const int gx0 = tx * TILE - 7;
    const int gy0 = ty * TILE - 7;

    /* padded weight table: index d+15, d = k - n in [-15..27], zero outside [0,12] */
    for (int i = tid; i < 48; i += 256) {
        int d = i - 15;
        s_gp[i] = ((unsigned)d < 13u) ? gauss[d] : 0.0f;
    }
    __syncthreads();

    /* per-lane weight fragments, shared by hblur-B (Wh[k][n]=g[k-n]) and
       vblur-A (Wv[m][k]=g[k-m]); invariant across tiles and channels */
    v2f wf[7];
#pragma unroll
    for (int kb = 0; kb < 7; ++kb) {
        const int k0 = kb * 4 + khalf * 2;
        wf[kb].x = s_gp[k0 + 0 - mrow + 15];
        wf[kb].y = s_gp[k0 + 1 - mrow + 15];
    }

    float accx[9], accy[9], accm[9];
#pragma unroll
    for (int p = 0; p < 9; ++p) { accx[p] = 0.0f; accy[p] = 0.0f; accm[p] = 0.0f; }

    const bool interior = (gx0 >= 0) && (gy0 >= 0) &&
                          (gx0 + IND <= Wimg) && (gy0 + IND <= Himg);

    for (int ch = 0; ch < NCH; ++ch) {
        const float* src = img + ((size_t)b * NCH + ch) * HWimg;

        /* ---- load 80x80 input region into LDS ---- */
#if HAVE_ASYNC
        if (interior) {
            for (int i = tid; i < IND * IND; i += 256) {
                int r = i / IND, c = i - r * IND;
                ASYNC_CP(&s_in[r * INS + c],
                         src + (size_t)(gy0 + r) * Wimg + (gx0 + c));
            }
        } else
#endif
        {
            for (int i = tid; i < IND * IND; i += 256) {
                int r = i / IND, c = i - r * IND;
                int y = gy0 + r, x = gx0 + c;
                float v = 0.0f;
                if ((unsigned)y < (unsigned)Himg && (unsigned)x < (unsigned)Wimg)
                    v = src[y * Wimg + x];
                s_in[r * INS + c] = v;
            }
        }
        /* prefetch next channel's tile (one per 128B line) */
        if (ch + 1 < NCH) {
            for (int q = tid; q < IND * 3; q += 256) {
                int r = q % IND, seg = q / IND;
                int y = gy0 + r, x = gx0 + seg * 32;
                if ((unsigned)y < (unsigned)Himg && (unsigned)x < (unsigned)Wimg)
                    __builtin_prefetch(src + HWimg + (size_t)y * Wimg + x, 0, 0);
            }
        }
#if HAVE_ASYNC
        if (interior) WAIT_ASYNC();
#endif
        __syncthreads();

        /* ---- horizontal blur: 5x4 tiles; Out[16x16] = In[16x28] x Wh[28x16] ---- */
#pragma unroll
        for (int rep = 0; rep < 3; ++rep) {
            const int t = wave + rep * 8;
            if (t < 20) {
                const int r0 = (t >> 2) * 16, c0 = (t & 3) * 16;
                v8f acc = {};
#pragma unroll
                for (int kb = 0; kb < 7; ++kb) {
                    const int k0 = kb * 4 + khalf * 2;
                    v2f a;
                    a.x = s_in[(r0 + mrow) * INS + c0 + k0 + 0];
                    a.y = s_in[(r0 + mrow) * INS + c0 + k0 + 1];
                    acc = __builtin_amdgcn_wmma_f32_16x16x4_f32(
                        false, a, false, wf[kb], (short)0, acc, false, false);
                }
#pragma unroll
                for (int i = 0; i < 8; ++i)
                    s_hb[(r0 + i + khalf * 8) * HBS + c0 + mrow] = acc[i];
            }
        }
        __syncthreads();

        /* ---- vertical blur: 4x4 tiles; Out[16x16] = Wv[16x28] x Hb[28x16] ---- */
#pragma unroll
        for (int rep = 0; rep < 2; ++rep) {
            const int t = wave + rep * 8;
            const int r0 = (t >> 2) * 16, c0 = (t & 3) * 16;
            v8f acc = {};
#pragma unroll
            for (int kb = 0; kb < 7; ++kb) {
                const int k0 = kb * 4 + khalf * 2;
                v2f bb;
                bb.x = s_hb[(r0 + k0 + 0) * HBS + c0 + mrow];
                bb.y = s_hb[(r0 + k0 + 1) * HBS + c0 + mrow];
                acc = __builtin_amdgcn_wmma_f32_16x16x4_f32(
                    false, wf[kb], false, bb, (short)0, acc, false, false);
            }
#pragma unroll
            for (int i = 0; i < 8; ++i)
                s_bl[(r0 + i + khalf * 8) * HBS + c0 + mrow] = acc[i];
        }
        __syncthreads();

        /* ---- Sobel over 48x48 outputs (9 px/thread); OOB blurred samples = 0 ---- */
        auto blv = [&](int rr, int cc) -> float {
            int y = ty * TILE - 1 + rr, x = tx * TILE - 1 + cc;
            if ((unsigned)y >= (unsigned)Himg || (unsigned)x >= (unsigned)Wimg) return 0.0f;
            return s_bl[rr * HBS + cc];
        };
#pragma unroll
        for (int p = 0; p < 9; ++p) {
            int idx = tid + p * 256;
            int py = idx / TILE, px = idx - py * TILE;
            int rr = py + 1, cc = px + 1;
            float tl = blv(rr - 1, cc - 1), tc = blv(rr - 1, cc), tr = blv(rr - 1, cc + 1);
            float ml = blv(rr, cc - 1),                           mr = blv(rr, cc + 1);
            float dl = blv(rr + 1, cc - 1), dc = blv(rr + 1, cc), dr = blv(rr + 1, cc + 1);
            float gx = (tl - tr) + 2.0f * (ml - mr) + (dl - dr);
            float gy = (tl + 2.0f * tc + tr) - (dl + 2.0f * dc + dr);
            accx[p] += gx;
            accy[p] += gy;
            accm[p] += sqrtf(gx * gx + gy * gy);
        }
        __syncthreads();
    }

#pragma unroll
    for (int p = 0; p < 9; ++p) {
        int idx = tid + p * 256;
        int py = idx / TILE, px = idx - py * TILE;
        int y = ty * TILE + py, x = tx * TILE + px;
        if (y < Himg && x < Wimg) {
            size_t o = (size_t)b * HWimg + (size_t)y * Wimg + x;
            gmag[o] = accm[p];
            float ang = atan2f(accy[p], accx[p]) * 57.29577951308232f + 180.0f;
            int ir = (int)rintf(ang * (1.0f / 45.0f));   /* RNE, like jnp.round */
            dirm[o] = (unsigned char)(ir & 7);
        }
    }
}

// ---------------------------------------------------------------------------
// NMS + double threshold -> code map {0: below, 1: middle, 2: strong}
// ---------------------------------------------------------------------------
__global__ __launch_bounds__(256)
void nms_kernel(const float* __restrict__ gmag, const unsigned char* __restrict__ dirm,
                const unsigned* __restrict__ info, unsigned char* __restrict__ code) {
    const float lowT  = ((const float*)info)[4];
    const float highT = ((const float*)info)[5];
    int i = blockIdx.x * 256 + threadIdx.x;
    if (i >= NPIX) return;
    int b = i >> 20;
    int p = i & (HWimg - 1);
    int y = p >> 10, x = p & (Wimg - 1);
    const float* m = gmag + (size_t)b * HWimg;
    float mc = m[p];
    const int dy[8] = {0, 1, 1, 1, 0, -1, -1, -1};
    const int dx[8] = {1, 1, 0, -1, -1, -1, 0, 1};
    int d = dirm[i];
    auto nbv = [&](int dd) -> float {
        int yy = y + dy[dd], xx = x + dx[dd];
        if ((unsigned)yy >= (unsigned)Himg || (unsigned)xx >= (unsigned)Wimg) return 0.0f;
        return m[yy * Wimg + xx];
    };
    float cp = mc - nbv(d);
    float cn = mc - nbv((d + 4) & 7);
    float thin = (fminf(cp, cn) > 0.0f) ? mc : 0.0f;
    unsigned char c = 0;
    if (thin > highT)                         c = 2;
    else if (thin >= lowT && thin <= highT)   c = 1;
    code[i] = c;
}

// ---------------------------------------------------------------------------
// Hysteresis: strong -> 1; middle with any strong 8-neighbor -> 1; borders 0
// ---------------------------------------------------------------------------
__global__ __launch_bounds__(256)
void hyst_kernel(const unsigned char* __restrict__ code, float* __restrict__ out) {
    int i = blockIdx.x * 256 + threadIdx.x;
    if (i >= NPIX) return;
    int b = i >> 20;
    int p = i & (HWimg - 1);
    int y = p >> 10, x = p & (Wimg - 1);
    float v = 0.0f;
    if (y > 0 && y < Himg - 1 && x > 0 && x < Wimg - 1) {
        const unsigned char* c = code + (size_t)b * HWimg;
        unsigned char cc = c[p];
        if (cc == 2) v = 1.0f;
        else if (cc == 1) {
            int hit = 0;
#pragma unroll
            for (int ddy = -1; ddy <= 1; ++ddy)
#pragma unroll
                for (int ddx = -1; ddx <= 1; ++ddx) {
                    if (ddy == 0 && ddx == 0) continue;
                    hit |= (c[(y + ddy) * Wimg + (x + ddx)] == 2);
                }
            if (hit) v = 1.0f;
        }
    }
    out[i] = v;
}

// ---------------------------------------------------------------------------
extern "C" void kernel_launch(void* const* d_in, const int* in_sizes, int n_in,
                              void* d_out, int out_size, void* d_ws, size_t ws_size,
                              hipStream_t stream) {
    const float* img   = (const float*)d_in[0];
    const float* gauss = (const float*)d_in[1];   /* 13-tap unnormalized gaussian */

    unsigned char* w = (unsigned char*)d_ws;
    unsigned* hist1 = (unsigned*)w;               /* 65536 u32 = 256 KB */
    unsigned* hist2 = hist1 + 65536;              /* 65536 u32 = 256 KB */
    unsigned* info  = hist2 + 65536;              /* 16 u32             */
    float*          gmag  = (float*)(w + (1u << 20));               /* 16 MB */
    unsigned char*  dirm  = w + (1u << 20) + (size_t)NPIX * 4;      /*  4 MB */
    unsigned char*  codem = dirm + (size_t)NPIX;                    /*  4 MB */
    float* out = (float*)d_out;

    (void)hipMemsetAsync(d_ws, 0, (size_t)2 * 65536 * sizeof(unsigned) + 64, stream);

    hist1_kernel<<<2048, 256, 0, stream>>>((const float4*)img, hist1, NTOT / 4);
    select_kernel<<<1, 1024, 0, stream>>>(hist1, info, 0);
    hist2_kernel<<<2048, 256, 0, stream>>>((const float4*)img, info, hist2, NTOT / 4);
    select_kernel<<<1, 1024, 0, stream>>>(hist2, info, 1);

    dim3 g((Wimg + TILE - 1) / TILE, (Himg + TILE - 1) / TILE, NBATCH);
    grad_kernel<<<g, 256, 0, stream>>>(img, gauss, gmag, dirm);

    nms_kernel<<<NPIX / 256, 256, 0, stream>>>(gmag, dirm, info, codem);
    hyst_kernel<<<NPIX / 256, 256, 0, stream>>>(codem, out);

    (void)in_sizes; (void)n_in; (void)out_size; (void)ws_size;
}